// GNN3_79783312490854
// MI455X (gfx1250) — compile-verified
//
#include <hip/hip_runtime.h>
#include <hip/hip_bf16.h>

typedef __attribute__((ext_vector_type(2))) float v2f;
typedef __attribute__((ext_vector_type(8))) float v8f;

#define N_DRUG 572
#define D      1024
#define S      64
#define KMAX   32
#define K2     2048           // 2*D
#define BN_EPS 1e-5f

// ---------------------------------------------------------------------------
// gfx1250 async global->LDS path (ASYNCcnt-tracked), with safe fallback.
// Probe result: builtin exists; param0 is `int __vector(4) addrspace(1)*`.
// ---------------------------------------------------------------------------
#if defined(__AMDGCN__) && __has_builtin(__builtin_amdgcn_global_load_async_to_lds_b128)
#define ASYNC_LDS 1
#else
#define ASYNC_LDS 0
#endif

__device__ __forceinline__ void copy16_to_lds(const float* g, float* l)
{
#if ASYNC_LDS
    // global_load_async_to_lds_b128: LDS dst from lane VGPR, no VGPR data path.
    typedef int v4i __attribute__((vector_size(16)));
    const v4i* gv = reinterpret_cast<const v4i*>(g);
    v4i*       lv = reinterpret_cast<v4i*>(l);
    __builtin_amdgcn_global_load_async_to_lds_b128(
        (__attribute__((address_space(1))) v4i*)gv,
        (__attribute__((address_space(3))) v4i*)lv,
        /*imm offset*/0, /*cpol*/0);
#else
    *reinterpret_cast<float4*>(l) = *reinterpret_cast<const float4*>(g);
#endif
}

__device__ __forceinline__ void wait_async_all()
{
#if ASYNC_LDS
#if __has_builtin(__builtin_amdgcn_s_wait_asynccnt)
    __builtin_amdgcn_s_wait_asynccnt(0);
#else
    asm volatile("s_wait_asynccnt 0" ::: "memory");
#endif
#endif
}

// ---------------------------------------------------------------------------
// Kernel 1: dot-product attention over neighbors (memory-bound gather phase).
// One block per drug node; query row staged in LDS; one wave per neighbor s
// for the score dot-products; block-wide softmax; then coalesced weighted
// gather of entity rows for `attended`.
// ---------------------------------------------------------------------------
__global__ __launch_bounds__(256) void attn_kernel(
    const float* __restrict__ drug_w, const float* __restrict__ rela_w,
    const float* __restrict__ ent_w,  const int* __restrict__ drug_name,
    const int* __restrict__ adj_tail, const int* __restrict__ adj_rel,
    float* __restrict__ attended)
{
    __shared__ float de[D];
    __shared__ float sc[S];
    __shared__ int   tails[S];
    __shared__ float inv_sum;

    const int n   = blockIdx.x;
    const int tid = threadIdx.x;
    const int row = drug_name[n];

    for (int d = tid; d < D; d += 256) de[d] = drug_w[(size_t)row * D + d];
    if (tid < S) tails[tid] = adj_tail[n * S + tid];
    __syncthreads();

    const int wave = tid >> 5;   // wave32 (gfx1250)
    const int lane = tid & 31;

    // scores[s] = <drug_e[n], rela_e[n,s]>
    for (int s = wave; s < S; s += 8) {
        const int rel = adj_rel[n * S + s];
        const float* rp = rela_w + (size_t)rel * D;
        float p = 0.f;
        for (int d = lane; d < D; d += 32) p += de[d] * rp[d];
#pragma unroll
        for (int off = 16; off > 0; off >>= 1) p += __shfl_xor(p, off, 32);
        if (lane == 0) sc[s] = p;
    }
    __syncthreads();

    // softmax over S=64 (tiny; serialized max/sum on one lane)
    if (tid == 0) {
        float m = sc[0];
        for (int s = 1; s < S; ++s) m = fmaxf(m, sc[s]);
        float sum = 0.f;
        for (int s = 0; s < S; ++s) { float e = __expf(sc[s] - m); sc[s] = e; sum += e; }
        inv_sum = 1.f / sum;
    }
    __syncthreads();
    if (tid < S) sc[tid] *= inv_sum;
    __syncthreads();

    // attended[n,d] = sum_s sc[s] * ent[tails[s], d]   (coalesced across d)
    for (int d = tid; d < D; d += 256) {
        float acc = 0.f;
#pragma unroll 4
        for (int s = 0; s < S; ++s)
            acc += sc[s] * ent_w[(size_t)tails[s] * D + d];
        attended[(size_t)n * D + d] = acc;
    }
}

// ---------------------------------------------------------------------------
// Kernel 2: h = relu( [attended | drug_e] @ lin_w^T + lin_b ) via
// V_WMMA_F32_16X16X4_F32 with double-buffered LDS tiles. Block = 8 waves ->
// 16(M) x 128(N) strip; K chunk 64. W tile (32 KB/step) is staged with
// GLOBAL_LOAD_ASYNC_TO_LDS_B128 (ASYNCcnt), overlapping the 16 WMMAs of the
// current chunk; one s_wait_asynccnt 0 + barrier per iteration. Rows padded
// to 68 floats: 16B-aligned rows, conflict-free strided fragment reads
// (ds_load_2addr_b64). A/B lane layout: lanes 0-15 carry K,K+1, lanes 16-31
// carry K+2,K+3.
// ---------------------------------------------------------------------------
#define KC    64
#define XLD   68
#define WLD   68

__global__ __launch_bounds__(256) void gemm_kernel(
    const float* __restrict__ attended, const float* __restrict__ drug_w,
    const int* __restrict__ drug_name,  const float* __restrict__ lin_w,
    const float* __restrict__ lin_b,    float* __restrict__ h)
{
    __shared__ float Xs[2][16 * XLD];
    __shared__ float Ws[2][128 * WLD];

    const int tid   = threadIdx.x;
    const int mBase = blockIdx.x * 16;
    const int nBase = blockIdx.y * 128;
    const int wave  = tid >> 5;
    const int lane  = tid & 31;
    const int nSub  = wave * 16;
    const int rsel  = lane & 15;          // M row (A) / N col (B) within tile
    const int koff  = (lane >> 4) << 1;   // lanes 16-31 hold K+2,K+3

    // per-thread staging coordinates
    const int xr   = tid >> 4;            // X row 0..15
    const int xc4  = (tid & 15) << 2;     // X col (float4)
    const int m_ld = mBase + xr;

    auto stage_tile = [&](int buf, int kBase) {
        // X tile: 16 x 64, one float4/thread (handles concat + row guard)
        {
            const int k = kBase + xc4;    // whole 64-chunk on one side of D
            float4 v = make_float4(0.f, 0.f, 0.f, 0.f);
            if (m_ld < N_DRUG) {
                const float* src = (k < D)
                    ? (attended + (size_t)m_ld * D + k)
                    : (drug_w + (size_t)drug_name[m_ld] * D + (k - D));
                v = *reinterpret_cast<const float4*>(src);
            }
            *reinterpret_cast<float4*>(&Xs[buf][xr * XLD + xc4]) = v;
        }
        // W tile: 128 x 64, 8 async 16B copies/thread
#pragma unroll
        for (int j = 0; j < 8; ++j) {
            const int vidx = tid + j * 256;
            const int nn   = vidx >> 4;
            const int cc   = (vidx & 15) << 2;
            copy16_to_lds(lin_w + (size_t)(nBase + nn) * K2 + kBase + cc,
                          &Ws[buf][nn * WLD + cc]);
        }
    };

    v8f acc = {};

    stage_tile(0, 0);
    wait_async_all();
    __syncthreads();

    for (int kBase = 0; kBase < K2; kBase += KC) {
        const int  cur  = (kBase >> 6) & 1;
        const bool more = (kBase + KC) < K2;
        if (more) stage_tile(cur ^ 1, kBase + KC);   // overlaps compute below

        const float* xp = &Xs[cur][rsel * XLD + koff];
        const float* wp = &Ws[cur][(nSub + rsel) * WLD + koff];
#pragma unroll
        for (int kc = 0; kc < KC; kc += 4) {
            v2f a, b;
            a.x = xp[kc]; a.y = xp[kc + 1];
            b.x = wp[kc]; b.y = wp[kc + 1];
            acc = __builtin_amdgcn_wmma_f32_16x16x4_f32(
                false, a, false, b, (short)0, acc, false, false);
        }

        if (more) {
            wait_async_all();   // this wave's async fills of cur^1 done
            __syncthreads();    // all waves done reading cur + fills visible
        }
    }

    // ---- epilogue: bias + ReLU, C layout: VGPR r -> M=r (lanes 0-15) / M=r+8 ----
    const int nG   = nBase + nSub + rsel;
    const int mOff = (lane >> 4) << 3;
    const float bias = lin_b[nG];
#pragma unroll
    for (int r = 0; r < 8; ++r) {
        const int m = mBase + r + mOff;
        if (m < N_DRUG) {
            float v = acc[r] + bias;
            h[(size_t)m * D + nG] = fmaxf(v, 0.f);
        }
    }
}

// ---------------------------------------------------------------------------
// Kernel 3: per-column batch statistics -> fused scale/shift.
// ---------------------------------------------------------------------------
__global__ __launch_bounds__(256) void bnstats_kernel(
    const float* __restrict__ h, const float* __restrict__ gamma,
    const float* __restrict__ beta, float* __restrict__ scale,
    float* __restrict__ shift)
{
    const int c = blockIdx.x * 256 + threadIdx.x;
    if (c >= D) return;
    float s = 0.f, q = 0.f;
    for (int r = 0; r < N_DRUG; ++r) {
        const float v = h[(size_t)r * D + c];
        s += v; q += v * v;
    }
    const float mean = s * (1.f / N_DRUG);
    const float var  = q * (1.f / N_DRUG) - mean * mean;
    const float sc   = gamma[c] * rsqrtf(var + BN_EPS);
    scale[c] = sc;
    shift[c] = beta[c] - mean * sc;
}

__global__ __launch_bounds__(256) void bnapply_kernel(
    float* __restrict__ out, const float* __restrict__ scale,
    const float* __restrict__ shift)
{
    const int i = blockIdx.x * 256 + threadIdx.x;
    if (i < N_DRUG * D) {
        const int c = i & (D - 1);
        out[i] = out[i] * scale[c] + shift[c];
    }
}

// ---------------------------------------------------------------------------
// Kernel 4: sequential neighbor smoothing. The scan dependency is row->row
// within a column; columns are independent, so thread = column replays the
// exact in-place update order. Working set is 2.3 MB (L2-resident at 192 MB);
// row reads are coalesced across threads.
// ---------------------------------------------------------------------------
__global__ __launch_bounds__(256) void smooth_kernel(
    float* __restrict__ out, const int* __restrict__ nbr_pos,
    const int* __restrict__ nbr_len, const int* __restrict__ epoch)
{
    if (epoch[0] <= 1) return;
    const int c = blockIdx.x * 256 + threadIdx.x;
    if (c >= D) return;
    for (int i = 0; i < N_DRUG; ++i) {
        const int len = nbr_len[i];          // uniform -> scalar loads
        if (len > 0) {
            float s = 0.f;
            for (int k = 0; k < len; ++k)
                s += out[(size_t)nbr_pos[i * KMAX + k] * D + c];
            const float cur = out[(size_t)i * D + c];
            out[(size_t)i * D + c] = (s / (float)len + cur) * 0.5f;
        }
    }
}

// ---------------------------------------------------------------------------
extern "C" void kernel_launch(void* const* d_in, const int* in_sizes, int n_in,
                              void* d_out, int out_size, void* d_ws, size_t ws_size,
                              hipStream_t stream)
{
    const float* drug_w   = (const float*)d_in[0];
    const float* rela_w   = (const float*)d_in[1];
    const float* ent_w    = (const float*)d_in[2];
    const float* lin_w    = (const float*)d_in[3];
    const float* lin_b    = (const float*)d_in[4];
    const float* gamma    = (const float*)d_in[5];
    const float* beta     = (const float*)d_in[6];
    const int*   drug_nm  = (const int*)d_in[7];
    const int*   adj_tail = (const int*)d_in[8];
    const int*   adj_rel  = (const int*)d_in[9];
    const int*   nbr_pos  = (const int*)d_in[10];
    const int*   nbr_len  = (const int*)d_in[11];
    const int*   epoch    = (const int*)d_in[12];

    float* out      = (float*)d_out;
    float* attended = (float*)d_ws;                       // 572*1024 f32
    float* scale    = attended + (size_t)N_DRUG * D;      // 1024 f32
    float* shift    = scale + D;                          // 1024 f32

    attn_kernel<<<N_DRUG, 256, 0, stream>>>(drug_w, rela_w, ent_w, drug_nm,
                                            adj_tail, adj_rel, attended);

    dim3 gg((N_DRUG + 15) / 16, D / 128);                 // 36 x 8
    gemm_kernel<<<gg, 256, 0, stream>>>(attended, drug_w, drug_nm,
                                        lin_w, lin_b, out);

    bnstats_kernel<<<(D + 255) / 256, 256, 0, stream>>>(out, gamma, beta, scale, shift);
    bnapply_kernel<<<(N_DRUG * D + 255) / 256, 256, 0, stream>>>(out, scale, shift);
    smooth_kernel<<<(D + 255) / 256, 256, 0, stream>>>(out, nbr_pos, nbr_len, epoch);
}